// GATModel_58798102282555
// MI455X (gfx1250) — compile-verified
//
#include <hip/hip_runtime.h>
#include <hip/hip_bf16.h>
#include <cstdint>

typedef __attribute__((ext_vector_type(2))) float v2f;
typedef __attribute__((ext_vector_type(8))) float v8f;

#define NEG_SLOPE 0.2f
#define ORD_NEG_INF 0x007FFFFFu   // f2ord(-inf)

// Order-preserving float <-> uint mapping so atomicMax(uint) == float max.
__device__ __forceinline__ unsigned f2ord(float f) {
    unsigned u = __float_as_uint(f);
    return (u & 0x80000000u) ? ~u : (u | 0x80000000u);
}
__device__ __forceinline__ float ord2f(unsigned o) {
    return __uint_as_float((o & 0x80000000u) ? (o & 0x7FFFFFFFu) : ~o);
}
__device__ __forceinline__ float lrelu(float v) { return v > 0.f ? v : NEG_SLOPE * v; }

// ---------------------------------------------------------------------------
// Kernel 1: h = x @ W1 via V_WMMA_F32_16X16X4_F32, fused per-head attention
// coefficients al_src/al_dst computed from the in-register accumulators.
// One wave => 16 rows x 64 cols (= one full head). 32 k-steps of K=4.
// ---------------------------------------------------------------------------
__global__ void k_gemm_att(const float* __restrict__ x, const float* __restrict__ W1,
                           const float* __restrict__ as1, const float* __restrict__ ad1,
                           float* __restrict__ h, float* __restrict__ als,
                           float* __restrict__ ald, int N) {
    const int wib  = threadIdx.x >> 5;
    const int lane = threadIdx.x & 31;
    const int tile = blockIdx.x * (blockDim.x >> 5) + wib;
    const int totalTiles = (N >> 4) * 4;          // row-tiles * heads
    if (tile >= totalTiles) return;               // wave-uniform guard (EXEC all-1s inside)

    const int m0   = (tile >> 2) << 4;            // row base
    const int g    = tile & 3;                    // head
    const int n0   = g << 6;                      // col base = g*64
    const int half = lane >> 4;                   // 0: lanes 0-15, 1: lanes 16-31
    const int l16  = lane & 15;

    v8f acc[4] = {};                              // four 16x16 f32 C/D tiles

    // A layout (16x4 f32): VGPR0 = K=0 (lanes0-15) / K=2 (lanes16-31), VGPR1 = K=1/K=3.
    // B layout (4x16 f32): same K split; N = lane%16.
    const float* xrow = x + (size_t)(m0 + l16) * 128;
    for (int k = 0; k < 128; k += 4) {
        v2f a;
        a.x = xrow[k + 2 * half];
        a.y = xrow[k + 2 * half + 1];
        const float* brow0 = W1 + (size_t)(k + 2 * half) * 256 + n0 + l16;
        const float* brow1 = W1 + (size_t)(k + 2 * half + 1) * 256 + n0 + l16;
#pragma unroll
        for (int nt = 0; nt < 4; ++nt) {
            v2f b;
            b.x = brow0[nt * 16];
            b.y = brow1[nt * 16];
            acc[nt] = __builtin_amdgcn_wmma_f32_16x16x4_f32(
                false, a, false, b, (short)0, acc[nt], false, false);
        }
    }

    // attention vectors for this head, per lane column
    float asv[4], adv[4];
#pragma unroll
    for (int nt = 0; nt < 4; ++nt) {
        asv[nt] = as1[n0 + nt * 16 + l16];
        adv[nt] = ad1[n0 + nt * 16 + l16];
    }

    // C/D layout: VGPR r holds row M = r (lanes 0-15) or r+8 (lanes 16-31), N = lane%16.
#pragma unroll
    for (int r = 0; r < 8; ++r) {
        const int row = m0 + r + 8 * half;
        float ss = 0.f, sd = 0.f;
#pragma unroll
        for (int nt = 0; nt < 4; ++nt) {
            float v = acc[nt][r];
            h[(size_t)row * 256 + n0 + nt * 16 + l16] = v;
            ss += v * asv[nt];
            sd += v * adv[nt];
        }
        // reduce across the 16 lanes of this half
#pragma unroll
        for (int off = 8; off; off >>= 1) {
            ss += __shfl_xor(ss, off, 16);
            sd += __shfl_xor(sd, off, 16);
        }
        if (l16 == 0) {
            als[row * 4 + g] = ss;
            ald[row * 4 + g] = sd;
        }
    }
}

// ---------------------------------------------------------------------------
// Init: segment-max buffers to ord(-inf)  (zeros handled by hipMemsetAsync)
// ---------------------------------------------------------------------------
__global__ void k_fill_ord(unsigned* __restrict__ p, int n) {
    int i = blockIdx.x * blockDim.x + threadIdx.x;
    if (i < n) p[i] = ORD_NEG_INF;
}

// ---------------------------------------------------------------------------
// Layer-1 edge pass A: e = lrelu(al_s[src] + al_d[dst]); segment max into mord.
// One thread per (edge, head). Edges >= E are the implicit self-loops.
// ---------------------------------------------------------------------------
__global__ void k_edge1a(const int* __restrict__ ei, int E, int N,
                         const float* __restrict__ als, const float* __restrict__ ald,
                         float* __restrict__ ebuf, unsigned* __restrict__ mord) {
    int t = blockIdx.x * blockDim.x + threadIdx.x;
    int Etot = E + N;
    if (t >= Etot * 4) return;
    int e = t >> 2, hd = t & 3;
    int s, d;
    if (e < E) { s = ei[e]; d = ei[E + e]; } else { s = d = e - E; }
    float v = lrelu(als[s * 4 + hd] + ald[d * 4 + hd]);
    ebuf[t] = v;
    atomicMax(&mord[d * 4 + hd], f2ord(v));
}

// ---------------------------------------------------------------------------
// Layer-1 edge pass B: ee = exp(e - m[dst]); denom[dst] += ee;
// agg[dst, head, :] += ee * h[src, head, :].  One wave per (edge, head),
// 2 channels per lane (float2), global f32 atomic adds.
// ---------------------------------------------------------------------------
__global__ void k_edge1b(const int* __restrict__ ei, int E, int N,
                         const float* __restrict__ ebuf, const unsigned* __restrict__ mord,
                         const float* __restrict__ h, float* __restrict__ denom,
                         float* __restrict__ agg) {
    int w = (blockIdx.x * blockDim.x + threadIdx.x) >> 5;
    int lane = threadIdx.x & 31;
    int Etot = E + N;
    if (w >= Etot * 4) return;
    int e = w >> 2, hd = w & 3;
    int s, d;
    if (e < E) { s = ei[e]; d = ei[E + e]; } else { s = d = e - E; }
    float ee = expf(ebuf[e * 4 + hd] - ord2f(mord[d * 4 + hd]));
    if (lane == 0) atomicAdd(&denom[d * 4 + hd], ee);
    int cbase = hd * 64 + lane * 2;
    const float* hs = h + (size_t)s * 256 + cbase;
    float* ag = agg + (size_t)d * 256 + cbase;
    atomicAdd(ag,     ee * hs[0]);
    atomicAdd(ag + 1, ee * hs[1]);
}

// ---------------------------------------------------------------------------
// Node pass: out1 = elu(agg/denom + b1) fused with layer-2 projection
// z[n] = out1[n,:] . W2[:,0].   One wave per node, 8 channels per lane.
// ---------------------------------------------------------------------------
__global__ void k_node1(const float* __restrict__ agg, const float* __restrict__ denom,
                        const float* __restrict__ b1, const float* __restrict__ W2,
                        float* __restrict__ z, int N) {
    int w = (blockIdx.x * blockDim.x + threadIdx.x) >> 5;
    int lane = threadIdx.x & 31;
    if (w >= N) return;
    const float* ag = agg + (size_t)w * 256;
    const float* dn = denom + w * 4;
    float acc = 0.f;
#pragma unroll
    for (int j = 0; j < 8; ++j) {
        int c = lane + 32 * j;
        float v = ag[c] / dn[c >> 6] + b1[c];
        v = v > 0.f ? v : expm1f(v);          // ELU (alpha = 1)
        acc += v * W2[(size_t)c * 64];        // W2[:, 0] of [256, 64]
    }
#pragma unroll
    for (int off = 16; off; off >>= 1) acc += __shfl_xor(acc, off);
    if (lane == 0) z[w] = acc;
}

// ---------------------------------------------------------------------------
// Layer-2 (heads=1, ch=1): scalar edge passes + fused numerator/denominator.
// ---------------------------------------------------------------------------
__global__ void k_edge2a(const int* __restrict__ ei, int E, int N,
                         const float* __restrict__ z,
                         const float* __restrict__ as2, const float* __restrict__ ad2,
                         float* __restrict__ e2buf, unsigned* __restrict__ m2ord) {
    int i = blockIdx.x * blockDim.x + threadIdx.x;
    int Etot = E + N;
    if (i >= Etot) return;
    int s, d;
    if (i < E) { s = ei[i]; d = ei[E + i]; } else { s = d = i - E; }
    float v = lrelu(z[s] * as2[0] + z[d] * ad2[0]);
    e2buf[i] = v;
    atomicMax(&m2ord[d], f2ord(v));
}

__global__ void k_edge2b(const int* __restrict__ ei, int E, int N,
                         const float* __restrict__ e2buf, const unsigned* __restrict__ m2ord,
                         const float* __restrict__ z,
                         float* __restrict__ den2, float* __restrict__ num2) {
    int i = blockIdx.x * blockDim.x + threadIdx.x;
    int Etot = E + N;
    if (i >= Etot) return;
    int s, d;
    if (i < E) { s = ei[i]; d = ei[E + i]; } else { s = d = i - E; }
    float ee = expf(e2buf[i] - ord2f(m2ord[d]));
    atomicAdd(&den2[d], ee);
    atomicAdd(&num2[d], ee * z[s]);
}

__global__ void k_node2(const float* __restrict__ num2, const float* __restrict__ den2,
                        const float* __restrict__ b2, float* __restrict__ out, int N) {
    int i = blockIdx.x * blockDim.x + threadIdx.x;
    if (i >= N) return;
    float v = num2[i] / den2[i] + b2[0];
    out[i] = 1.f / (1.f + expf(-v));
}

// ---------------------------------------------------------------------------
extern "C" void kernel_launch(void* const* d_in, const int* in_sizes, int n_in,
                              void* d_out, int out_size, void* d_ws, size_t ws_size,
                              hipStream_t stream) {
    (void)n_in; (void)out_size; (void)ws_size;
    const float* x   = (const float*)d_in[0];
    const int*   ei  = (const int*)  d_in[1];
    const float* W1  = (const float*)d_in[2];
    const float* as1 = (const float*)d_in[3];
    const float* ad1 = (const float*)d_in[4];
    const float* b1  = (const float*)d_in[5];
    const float* W2  = (const float*)d_in[6];
    const float* as2 = (const float*)d_in[7];
    const float* ad2 = (const float*)d_in[8];
    const float* b2  = (const float*)d_in[9];
    float* out = (float*)d_out;

    const int N = in_sizes[0] / 128;
    const int E = in_sizes[1] / 2;
    const int Etot = E + N;

    // ---- workspace layout (float elements) ----
    float* base = (float*)d_ws;
    size_t off = 0;
    float* h     = base + off; off += (size_t)N * 256;
    float* als   = base + off; off += (size_t)N * 4;
    float* ald   = base + off; off += (size_t)N * 4;
    float* ebuf  = base + off; off += (size_t)Etot * 4;
    float* agg   = base + off; off += (size_t)N * 256;   // |
    float* denom = base + off; off += (size_t)N * 4;     // | contiguous zero region
    float* den2  = base + off; off += (size_t)N;         // |
    float* num2  = base + off; off += (size_t)N;         // |
    unsigned* m1ord = (unsigned*)(base + off); off += (size_t)N * 4;  // | contiguous
    unsigned* m2ord = (unsigned*)(base + off); off += (size_t)N;      // | ord(-inf) region
    float* z     = base + off; off += (size_t)N;
    float* e2buf = base + off; off += (size_t)Etot;

    const int B = 256;

    // init accumulators
    hipMemsetAsync(agg, 0, (size_t)(N * 256 + N * 4 + 2 * N) * sizeof(float), stream);
    {
        int n = N * 5;  // m1ord (4N) + m2ord (N), contiguous
        k_fill_ord<<<(n + B - 1) / B, B, 0, stream>>>(m1ord, n);
    }

    // layer-1 GEMM + attention coefficients (WMMA f32 16x16x4)
    {
        int waves = (N / 16) * 4;
        k_gemm_att<<<(waves + 7) / 8, B, 0, stream>>>(x, W1, as1, ad1, h, als, ald, N);
    }

    // layer-1 segment softmax + aggregate
    {
        int t = Etot * 4;
        k_edge1a<<<(t + B - 1) / B, B, 0, stream>>>(ei, E, N, als, ald, ebuf, m1ord);
        int waves = Etot * 4;
        k_edge1b<<<(waves + 7) / 8, B, 0, stream>>>(ei, E, N, ebuf, m1ord, h, denom, agg);
    }

    // node pass: divide + bias + ELU + layer-2 projection
    k_node1<<<(N + 7) / 8, B, 0, stream>>>(agg, denom, b1, W2, z, N);

    // layer-2 scalar attention
    k_edge2a<<<(Etot + B - 1) / B, B, 0, stream>>>(ei, E, N, z, as2, ad2, e2buf, m2ord);
    k_edge2b<<<(Etot + B - 1) / B, B, 0, stream>>>(ei, E, N, e2buf, m2ord, z, den2, num2);

    // final: divide + bias + sigmoid
    k_node2<<<(N + B - 1) / B, B, 0, stream>>>(num2, den2, b2, out, N);
}